// GraphConvolution_1838246003406
// MI455X (gfx1250) — compile-verified
//
#include <hip/hip_runtime.h>
#include <hip/hip_bf16.h>

typedef __attribute__((ext_vector_type(2))) float v2f;
typedef __attribute__((ext_vector_type(8))) float v8f;

#define D_IN  128
#define D_OUT 64

// ---------------------------------------------------------------------------
// Kernel 1: out[i, j] = bias[j]   (initialize accumulation target)
// ---------------------------------------------------------------------------
__global__ __launch_bounds__(256) void gcn_bias_init_kernel(
    float* __restrict__ out, const float* __restrict__ bias, int total) {
  int i = blockIdx.x * 256 + threadIdx.x;
  if (i < total) out[i] = bias[i & (D_OUT - 1)];
}

// ---------------------------------------------------------------------------
// Kernel 2: support = X @ W via V_WMMA_F32_16X16X4_F32.
// One wave -> one 16x16 output tile. 4 waves/block cover N=64.
// A frag (16x4 f32): lanes 0-15 hold M=lane, K={k,k+1}; lanes 16-31 K={k+2,k+3}
// B frag (4x16 f32): lanes 0-15 hold N=lane, K={k,k+1}; lanes 16-31 K={k+2,k+3}
// D (16x16 f32, 8 VGPRs): vgpr r -> M = r + (lane>=16 ? 8 : 0), N = lane&15
// ---------------------------------------------------------------------------
__global__ __launch_bounds__(128) void gcn_gemm_wmma_kernel(
    const float* __restrict__ X,   // [n_rows, 128]
    const float* __restrict__ W,   // [128, 64]
    float* __restrict__ S,         // [n_rows, 64]
    int n_rows) {
  const int lane = threadIdx.x & 31;
  const int wave = threadIdx.x >> 5;           // N tile: 0..3
  const int m0   = blockIdx.x * 16;
  const int n0   = wave * 16;
  const int ml   = lane & 15;                  // M (for A) / N (for B) within tile
  const int kb   = (lane >> 4) * 2;            // 0 for lanes 0-15, 2 for 16-31
  if (m0 >= n_rows) return;

  const float* __restrict__ xrow = X + (long long)(m0 + ml) * D_IN;
  const float* __restrict__ wcol = W + n0 + ml;

  v8f acc = {};
#pragma unroll 8
  for (int k = 0; k < D_IN; k += 4) {
    v2f a, b;
    a.x = xrow[k + kb];                        // contiguous pair -> global_load_b64
    a.y = xrow[k + kb + 1];
    b.x = wcol[(long long)(k + kb) * D_OUT];
    b.y = wcol[(long long)(k + kb + 1) * D_OUT];
    acc = __builtin_amdgcn_wmma_f32_16x16x4_f32(
        /*neg_a=*/false, a, /*neg_b=*/false, b,
        /*c_mod=*/(short)0, acc, /*reuse_a=*/false, /*reuse_b=*/false);
  }

  const int mhi = (lane >> 4) * 8;
#pragma unroll
  for (int r = 0; r < 8; ++r) {
    S[(long long)(m0 + mhi + r) * D_OUT + n0 + ml] = acc[r];
  }
}

// ---------------------------------------------------------------------------
// Kernel 3: edge scatter. 16 threads per edge, float4 gather (L2-resident
// support), 4x global_atomic_add_f32 (non-returning, L2 atomic units).
// ---------------------------------------------------------------------------
__global__ __launch_bounds__(256) void gcn_edge_kernel(
    const int*   __restrict__ adj_row,
    const int*   __restrict__ adj_col,
    const float* __restrict__ adj_val,
    const float* __restrict__ support,   // [n_nodes, 64]
    float*       __restrict__ out,       // [n_nodes, 64]
    int n_edges) {
  long long idx = (long long)blockIdx.x * 256 + threadIdx.x;
  int e = (int)(idx >> 4);
  if (e >= n_edges) return;
  int c = ((int)idx & 15) * 4;

  const int   r = adj_row[e];
  const int   s = adj_col[e];
  const float v = adj_val[e];

  const float4 sv = *reinterpret_cast<const float4*>(
      support + (long long)s * D_OUT + c);
  float* op = out + (long long)r * D_OUT + c;
  atomicAdd(op + 0, v * sv.x);
  atomicAdd(op + 1, v * sv.y);
  atomicAdd(op + 2, v * sv.z);
  atomicAdd(op + 3, v * sv.w);
}

// ---------------------------------------------------------------------------
// Launch
// ---------------------------------------------------------------------------
extern "C" void kernel_launch(void* const* d_in, const int* in_sizes, int n_in,
                              void* d_out, int out_size, void* d_ws, size_t ws_size,
                              hipStream_t stream) {
  const float* X    = (const float*)d_in[0];   // [N, 128]
  const float* W    = (const float*)d_in[1];   // [128, 64]
  const float* bias = (const float*)d_in[2];   // [64]
  const int*   row  = (const int*)  d_in[3];   // [E]
  const int*   col  = (const int*)  d_in[4];   // [E]
  const float* val  = (const float*)d_in[5];   // [E]
  float* out = (float*)d_out;                  // [N, 64]

  const int n_nodes = in_sizes[0] / D_IN;
  const int n_edges = in_sizes[3];

  float* support = (float*)d_ws;               // needs n_nodes*64*4 = 25.6 MB

  // 1) out = bias (broadcast)
  {
    int total  = n_nodes * D_OUT;
    int blocks = (total + 255) / 256;
    gcn_bias_init_kernel<<<blocks, 256, 0, stream>>>(out, bias, total);
  }

  // 2) support = X @ W  (WMMA f32 16x16x4)
  {
    int mtiles = (n_nodes + 15) / 16;          // 6250 for N=100000 (exact)
    gcn_gemm_wmma_kernel<<<mtiles, 128, 0, stream>>>(X, W, support, n_nodes);
  }

  // 3) out[row[e]] += val[e] * support[col[e]]
  {
    long long threads = (long long)n_edges * 16;
    int blocks = (int)((threads + 255) / 256);
    gcn_edge_kernel<<<blocks, 256, 0, stream>>>(row, col, val, support, out,
                                                n_edges);
  }
}